// DecGreenNet_product_CP2_90967407329571
// MI455X (gfx1250) — compile-verified
//
#include <hip/hip_runtime.h>
#include <cmath>

#define NPTS 16384
#define DIN  3
#define HID  1024
#define RR   64
#define KDIM 64
#define RK   4096   // R*K == R*R
#define NQ   4096

typedef __attribute__((ext_vector_type(2))) float v2f;
typedef __attribute__((ext_vector_type(8))) float v8f;

// gfx1250 has native V_TANH_F32 (TRANS op, co-executes with VALU FMA)
__device__ __forceinline__ float fast_tanh(float x) {
#if __has_builtin(__builtin_amdgcn_tanhf)
    return __builtin_amdgcn_tanhf(x);
#else
    return tanhf(x);
#endif
}

// agent-scope relaxed fp32 atomic add -> global_atomic_add_f32
__device__ __forceinline__ void atomAddF(float* p, float v) {
    __hip_atomic_fetch_add(p, v, __ATOMIC_RELAXED, __HIP_MEMORY_SCOPE_AGENT);
}

// ---------------------------------------------------------------------------
// init 1: zero the t0/t1/Y accumulators and w
// ws layout: t0[0,1024) t1[1024,2048) Y0=2048 Y1=2049 ... w[16384,17408)
// ---------------------------------------------------------------------------
__global__ void init_t_kernel(float* __restrict__ ws)
{
    const int idx = blockIdx.x * blockDim.x + threadIdx.x;
    if (idx < 2050)  ws[idx] = 0.0f;            // t0, t1, Y0, Y1
    if (idx < 1024)  ws[16384 + idx] = 0.0f;    // w
}

// ---------------------------------------------------------------------------
// Kernel 1: partial t[h] += sum_{a in chunk} y[a]*tanh(qx[a,:]·Wa[:,h]+ba[h])
//           y[a] = exp(-ep*||qx[a]||^2);  Y += sum y[a]
// grid (4 h-blocks, 8 row-chunks) x 256; each thread 512 rows of tanh.
// ---------------------------------------------------------------------------
__global__ void quad_reduce_kernel(const float* __restrict__ qx,
                                   const float* __restrict__ Wa,   // [3,HID]
                                   const float* __restrict__ ba,   // [HID]
                                   const float* __restrict__ eqp,  // [1]
                                   float* __restrict__ t_out,      // [HID]
                                   float* __restrict__ Y_out)      // [1]
{
    __shared__ float sq[256 * 3];
    __shared__ float sy[256];
    const int tid  = threadIdx.x;
    const int h    = blockIdx.x * blockDim.x + tid;     // 0..1023
    const int base = blockIdx.y * 512;                  // row chunk of 512
    const float ep = eqp[0];
    const float w0 = Wa[h];
    const float w1 = Wa[HID + h];
    const float w2 = Wa[2 * HID + h];
    const float bb = ba[h];
    float acc  = 0.0f;
    float ysum = 0.0f;
    for (int c = 0; c < 512; c += 256) {
        const int a = base + c + tid;
        const float x0 = qx[a * 3 + 0];
        const float x1 = qx[a * 3 + 1];
        const float x2 = qx[a * 3 + 2];
        sq[tid * 3 + 0] = x0;
        sq[tid * 3 + 1] = x1;
        sq[tid * 3 + 2] = x2;
        sy[tid] = __expf(-ep * (x0 * x0 + x1 * x1 + x2 * x2));
        __syncthreads();
        #pragma unroll 4
        for (int r = 0; r < 256; ++r) {
            float z = fmaf(w0, sq[r * 3 + 0],
                      fmaf(w1, sq[r * 3 + 1],
                      fmaf(w2, sq[r * 3 + 2], bb)));
            acc = fmaf(sy[r], fast_tanh(z), acc);
        }
        if (blockIdx.x == 0 && tid == 0) {
            for (int r = 0; r < 256; ++r) ysum += sy[r];
        }
        __syncthreads();
    }
    atomAddF(&t_out[h], acc);
    if (blockIdx.x == 0 && tid == 0) atomAddF(Y_out, ysum);
}

// ---------------------------------------------------------------------------
// init 2: S0[j] = Y0*bq0b[j]; S1[j] = Y1*bq1b[j]
// ---------------------------------------------------------------------------
__global__ void init_S_kernel(const float* __restrict__ bq0b,
                              const float* __restrict__ bq1b,
                              const float* __restrict__ Y0,
                              const float* __restrict__ Y1,
                              float* __restrict__ S0,
                              float* __restrict__ S1)
{
    const int j = blockIdx.x * blockDim.x + threadIdx.x;
    S0[j] = Y0[0] * bq0b[j];
    S1[j] = Y1[0] * bq1b[j];
}

// ---------------------------------------------------------------------------
// Kernel 2: S[j] += sum_{h in chunk} t[h]*M[h,j]   (M row-major [HID,RK])
// grid (16 col-blocks, 4 h-chunks of 256) x 256; coalesced, t chunk in LDS.
// ---------------------------------------------------------------------------
__global__ void gemv_cols_kernel(const float* __restrict__ t,   // [HID]
                                 const float* __restrict__ M,   // [HID,RK]
                                 float* __restrict__ S)         // [RK]
{
    __shared__ float st[256];
    const int tid = threadIdx.x;
    const int h0  = blockIdx.y * 256;
    st[tid] = t[h0 + tid];
    __syncthreads();
    const int j = blockIdx.x * blockDim.x + tid;
    float acc = 0.0f;
    for (int h = 0; h < 256; h += 8) {
        // speculative prefetch ahead on the streaming weight matrix
        __builtin_prefetch(&M[(size_t)(h0 + h + 32) * RK + j], 0, 0);
        #pragma unroll
        for (int u = 0; u < 8; ++u)
            acc = fmaf(st[h + u], M[(size_t)(h0 + h + u) * RK + j], acc);
    }
    atomAddF(&S[j], acc);
}

// ---------------------------------------------------------------------------
// Kernel 3: v = (S0 @ S1^T).flatten()  64x64x64 GEMM via V_WMMA_F32_16X16X4_F32
// one block, 512 threads = 16 waves; wave (tm,tn) owns one 16x16 C tile.
// ---------------------------------------------------------------------------
__global__ void rhs_wmma_kernel(const float* __restrict__ S0,   // [RR,KDIM]
                                const float* __restrict__ S1,   // [RR,KDIM]
                                float* __restrict__ v)          // [RR*RR]
{
    const int tid  = threadIdx.x;
    const int lane = tid & 31;
    const int wv   = tid >> 5;             // 0..15
    const int tm   = (wv >> 2) * 16;       // C row-tile base
    const int tn   = (wv & 3) * 16;        // C col-tile base
    const int half = lane >> 4;            // 0 or 1
    const int l16  = lane & 15;
    v8f c = {};
    for (int kt = 0; kt < KDIM; kt += 4) {
        const int kb = kt + half * 2;
        v2f a, b;
        a.x = S0[(tm + l16) * KDIM + kb];
        a.y = S0[(tm + l16) * KDIM + kb + 1];
        b.x = S1[(tn + l16) * KDIM + kb];      // B = S1^T : B[k][n] = S1[n][k]
        b.y = S1[(tn + l16) * KDIM + kb + 1];
        c = __builtin_amdgcn_wmma_f32_16x16x4_f32(
                /*neg_a=*/false, a, /*neg_b=*/false, b,
                /*c_mod=*/(short)0, c, /*reuse_a=*/false, /*reuse_b=*/false);
    }
    #pragma unroll
    for (int r = 0; r < 8; ++r) {
        const int m = tm + half * 8 + r;
        v[m * RR + (tn + l16)] = c[r];
    }
}

// ---------------------------------------------------------------------------
// Kernel 4: w[row] += Wx2[row, half]·v[half] (wave per (row,half)); c = bx2·v
// grid = 257 blocks x 256: blocks 0..255 -> 8 waves = 4 rows x 2 halves each;
// block 256 reduces c.
// ---------------------------------------------------------------------------
__global__ void gemv_rows_kernel(const float* __restrict__ Wx2, // [HID,RK]
                                 const float* __restrict__ bx2, // [RK]
                                 const float* __restrict__ v,   // [RK]
                                 float* __restrict__ w,         // [HID]
                                 float* __restrict__ cscal)     // [1]
{
    __shared__ float red[256];
    const int tid = threadIdx.x;
    if (blockIdx.x < 256) {
        const int lane = tid & 31;
        const int wave = tid >> 5;                 // 0..7
        const int glob = blockIdx.x * 8 + wave;    // 0..2047
        const int row  = glob >> 1;                // 0..1023
        const int half = glob & 1;
        const float* Mrow = Wx2 + (size_t)row * RK + half * 2048;
        const float* vh   = v + half * 2048;
        float acc = 0.0f;
        #pragma unroll 4
        for (int j = lane; j < 2048; j += 32)
            acc = fmaf(Mrow[j], vh[j], acc);
        #pragma unroll
        for (int off = 16; off > 0; off >>= 1)
            acc += __shfl_xor(acc, off, 32);
        if (lane == 0) atomAddF(&w[row], acc);
    } else {
        float acc = 0.0f;
        for (int j = tid; j < RK; j += 256)
            acc = fmaf(bx2[j], v[j], acc);
        red[tid] = acc;
        __syncthreads();
        for (int s = 128; s > 0; s >>= 1) {
            if (tid < s) red[tid] += red[tid + s];
            __syncthreads();
        }
        if (tid == 0) cscal[0] = red[0];
    }
}

// ---------------------------------------------------------------------------
// init 3: out[n] = c  (re-initialized every call -> replay-deterministic)
// ---------------------------------------------------------------------------
__global__ void init_out_kernel(const float* __restrict__ cs,
                                float* __restrict__ out)
{
    const int n = blockIdx.x * blockDim.x + threadIdx.x;
    out[n] = cs[0];
}

// ---------------------------------------------------------------------------
// Kernel 5: out[n] += sum_{h in chunk} tanh(x[n,:]·Wx1[:,h]+bx1[h]) * w[h]
// grid (64 n-blocks, 8 h-chunks of 128) x 256; 128 hardware-tanh per thread.
// ---------------------------------------------------------------------------
__global__ void final_kernel(const float* __restrict__ x,    // [NPTS,3]
                             const float* __restrict__ Wx1,  // [3,HID]
                             const float* __restrict__ bx1,  // [HID]
                             const float* __restrict__ w,    // [HID]
                             float* __restrict__ out)        // [NPTS]
{
    __shared__ float sW0[128], sW1[128], sW2[128], sB[128], sw[128];
    const int tid = threadIdx.x;
    const int h0  = blockIdx.y * 128;
    if (tid < 128) {
        sW0[tid] = Wx1[h0 + tid];
        sW1[tid] = Wx1[HID + h0 + tid];
        sW2[tid] = Wx1[2 * HID + h0 + tid];
        sB[tid]  = bx1[h0 + tid];
        sw[tid]  = w[h0 + tid];
    }
    __syncthreads();
    const int n = blockIdx.x * blockDim.x + tid;
    const float x0 = x[n * 3 + 0];
    const float x1 = x[n * 3 + 1];
    const float x2 = x[n * 3 + 2];
    float acc = 0.0f;
    #pragma unroll 4
    for (int h = 0; h < 128; ++h) {
        float z = fmaf(x0, sW0[h], fmaf(x1, sW1[h], fmaf(x2, sW2[h], sB[h])));
        acc = fmaf(fast_tanh(z), sw[h], acc);
    }
    atomAddF(&out[n], acc);
}

// ---------------------------------------------------------------------------
extern "C" void kernel_launch(void* const* d_in, const int* in_sizes, int n_in,
                              void* d_out, int out_size, void* d_ws, size_t ws_size,
                              hipStream_t stream)
{
    (void)in_sizes; (void)n_in; (void)out_size; (void)ws_size;
    const float* input = (const float*)d_in[0];
    const float* eqp   = (const float*)d_in[1];
    const float* qx0   = (const float*)d_in[2];
    const float* qx1   = (const float*)d_in[3];
    const float* Wx1   = (const float*)d_in[4];
    const float* bx1   = (const float*)d_in[5];
    const float* Wx2   = (const float*)d_in[6];
    const float* bx2   = (const float*)d_in[7];
    const float* Wq0a  = (const float*)d_in[8];
    const float* bq0a  = (const float*)d_in[9];
    const float* Wq0b  = (const float*)d_in[10];
    const float* bq0b  = (const float*)d_in[11];
    const float* Wq1a  = (const float*)d_in[12];
    const float* bq1a  = (const float*)d_in[13];
    const float* Wq1b  = (const float*)d_in[14];
    const float* bq1b  = (const float*)d_in[15];
    float* out = (float*)d_out;
    float* ws  = (float*)d_ws;

    float* t0 = ws;             // 1024
    float* t1 = ws + 1024;      // 1024
    float* Y0 = ws + 2048;      // 1
    float* Y1 = ws + 2049;      // 1
    float* S0 = ws + 4096;      // 4096
    float* S1 = ws + 8192;      // 4096
    float* v  = ws + 12288;     // 4096
    float* wv = ws + 16384;     // 1024
    float* cs = ws + 17408;     // 1

    init_t_kernel<<<9, 256, 0, stream>>>(ws);
    quad_reduce_kernel<<<dim3(4, 8), 256, 0, stream>>>(qx0, Wq0a, bq0a, eqp, t0, Y0);
    quad_reduce_kernel<<<dim3(4, 8), 256, 0, stream>>>(qx1, Wq1a, bq1a, eqp, t1, Y1);
    init_S_kernel<<<16, 256, 0, stream>>>(bq0b, bq1b, Y0, Y1, S0, S1);
    gemv_cols_kernel<<<dim3(16, 4), 256, 0, stream>>>(t0, Wq0b, S0);
    gemv_cols_kernel<<<dim3(16, 4), 256, 0, stream>>>(t1, Wq1b, S1);
    rhs_wmma_kernel<<<1, 512, 0, stream>>>(S0, S1, v);
    gemv_rows_kernel<<<257, 256, 0, stream>>>(Wx2, bx2, v, wv, cs);
    init_out_kernel<<<NPTS / 256, 256, 0, stream>>>(cs, out);
    final_kernel<<<dim3(64, 8), 256, 0, stream>>>(input, Wx1, bx1, wv, out);
}